// _RoIPooling_12369505813075
// MI455X (gfx1250) — compile-verified
//
#include <hip/hip_runtime.h>
#include <hip/hip_bf16.h>

// RoI max-pool for MI455X (gfx1250, wave32).
// features: (2, 512, 50, 75) f32 ; rois: (2048, 5) f32 ; out: (2048, 512, 7, 7) f32
// Memory-bound: 205 MB NT output stores (HBM), ~1.8 GB feature reads (L2-resident).

#define RP_C      512
#define RP_H      50
#define RP_W      75
#define RP_HW     (RP_H * RP_W)          // 3750
#define RP_BINS   49
#define RP_PER_ROI (RP_C * RP_BINS)      // 25088
#define RP_SCALE  0.0625f
#define RP_NEG    (-3.402823466e38f)

// ---- CDNA5 async-to-LDS availability probe (safe fallback if absent) ----
#if defined(__gfx1250__) && \
    __has_builtin(__builtin_amdgcn_global_load_async_to_lds_b128) && \
    __has_builtin(__builtin_amdgcn_s_wait_asynccnt)
#define RP_ASYNC 1
#else
#define RP_ASYNC 0
#if defined(__HIP_DEVICE_COMPILE__)
#warning "RP_ASYNC fallback in use: global_load_async_to_lds_b128 not available"
#endif
#endif

#if RP_ASYNC
typedef int rp_v4i __attribute__((vector_size(4 * sizeof(int))));
typedef __attribute__((address_space(1))) rp_v4i* rp_gptr;
typedef __attribute__((address_space(3))) rp_v4i* rp_lptr;
#endif

// Mirror the reference bin-bound math exactly:
//   x1 = rint(roi*scale) (RTNE, matches jnp.round)
//   hstart = (int)clip(floor(ph*bin_h) + y1, 0, H)   (truncation matches astype(int32))
__device__ __forceinline__ void rp_compute_bounds(const float* __restrict__ roi,
                                                  int& b, int* hb, int* wb) {
    b = (int)roi[0];
    float x1 = rintf(roi[1] * RP_SCALE);
    float y1 = rintf(roi[2] * RP_SCALE);
    float x2 = rintf(roi[3] * RP_SCALE);
    float y2 = rintf(roi[4] * RP_SCALE);
    float roi_w = fmaxf(x2 - x1 + 1.0f, 1.0f);
    float roi_h = fmaxf(y2 - y1 + 1.0f, 1.0f);
    float bin_h = roi_h / 7.0f;
    float bin_w = roi_w / 7.0f;
#pragma unroll
    for (int p = 0; p < 7; ++p) {
        float fp = (float)p;
        float hs = fminf(fmaxf(floorf(fp * bin_h) + y1, 0.0f), (float)RP_H);
        float he = fminf(fmaxf(ceilf((fp + 1.0f) * bin_h) + y1, 0.0f), (float)RP_H);
        float ws = fminf(fmaxf(floorf(fp * bin_w) + x1, 0.0f), (float)RP_W);
        float we = fminf(fmaxf(ceilf((fp + 1.0f) * bin_w) + x1, 0.0f), (float)RP_W);
        hb[p] = ((int)hs) | (((int)he) << 16);
        wb[p] = ((int)ws) | (((int)we) << 16);
    }
}

// Kernel A: one thread per ROI -> 16-int packed descriptor in workspace.
// layout per ROI: [0]=batch, [1..7]=h-bounds(packed), [8..14]=w-bounds(packed), [15]=pad
__global__ void roi_setup_kernel(const float* __restrict__ rois,
                                 int* __restrict__ bounds, int R) {
    int r = blockIdx.x * blockDim.x + threadIdx.x;
    if (r >= R) return;
    int b, hb[7], wb[7];
    rp_compute_bounds(rois + (size_t)r * 5, b, hb, wb);
    int* o = bounds + (size_t)r * 16;
    o[0] = b;
#pragma unroll
    for (int p = 0; p < 7; ++p) { o[1 + p] = hb[p]; o[8 + p] = wb[p]; }
    o[15] = 0;
}

// Kernel B: one thread per output element (r, c, ph, pw) — fully coalesced NT stores.
// A 256-thread block covers 256 consecutive outputs => at most 2 distinct ROIs;
// their 64-B descriptors are staged into LDS via async-to-LDS (ASYNCcnt path).
template <bool USE_WS>
__global__ __launch_bounds__(256)
void roi_pool_kernel(const float* __restrict__ feat,
                     const float* __restrict__ rois,
                     const int* __restrict__ bounds,
                     float* __restrict__ out, int total) {
    __shared__ int sdesc[32];
    const int tid   = threadIdx.x;
    const int first = blockIdx.x * 256;
    const int idx   = first + tid;
    const int lastI = (first + 255 < total - 1) ? (first + 255) : (total - 1);
    const int r0 = first / RP_PER_ROI;
    const int r1 = lastI / RP_PER_ROI;   // r1 == r0 or r0+1

    if (USE_WS) {
#if RP_ASYNC
        if (tid < 8) {
            int rr = (tid >> 2) ? r1 : r0;
            __builtin_amdgcn_global_load_async_to_lds_b128(
                (rp_gptr)(bounds + (size_t)rr * 16 + (tid & 3) * 4),
                (rp_lptr)(sdesc + tid * 4), 0, 0);
        }
        __builtin_amdgcn_s_wait_asynccnt(0);
        __syncthreads();
#else
        if (tid < 32) {
            int rr = (tid < 16) ? r0 : r1;
            sdesc[tid] = bounds[(size_t)rr * 16 + (tid & 15)];
        }
        __syncthreads();
#endif
    } else {
        if (tid < 2) {
            int rr = tid ? r1 : r0;
            int b, hb[7], wb[7];
            rp_compute_bounds(rois + (size_t)rr * 5, b, hb, wb);
            int* o = sdesc + tid * 16;
            o[0] = b;
#pragma unroll
            for (int p = 0; p < 7; ++p) { o[1 + p] = hb[p]; o[8 + p] = wb[p]; }
        }
        __syncthreads();
    }

    if (idx >= total) return;

    int r   = idx / RP_PER_ROI;
    int rem = idx - r * RP_PER_ROI;
    int c   = rem / RP_BINS;
    int bin = rem - c * RP_BINS;
    int ph  = bin / 7;
    int pw  = bin - ph * 7;

    const int* d = sdesc + ((r == r0) ? 0 : 16);
    int b   = d[0];
    int hbp = d[1 + ph];
    int wbp = d[8 + pw];
    int hs = hbp & 0xFFFF, he = hbp >> 16;
    int ws = wbp & 0xFFFF, we = wbp >> 16;

    const float* base = feat + (size_t)(b * RP_C + c) * RP_HW;
    const float* row  = base + hs * RP_W + ws;
    const int wcnt = we - ws;
    const bool valid = (hs < he) && (wcnt > 0);

    if (he - hs > 1) __builtin_prefetch(row + RP_W, 0, 0);  // global_prefetch_b8: next row

    float m = RP_NEG;
    for (int h = hs; h < he; ++h) {
        int w = 0;
        for (; w + 1 < wcnt; w += 2)                 // unroll x2: fewer branches
            m = fmaxf(m, fmaxf(row[w], row[w + 1]));
        if (w < wcnt)
            m = fmaxf(m, row[w]);
        row += RP_W;
    }

    // Non-temporal store: keep the (L2-resident) feature map from being evicted
    // by 205 MB of streaming output.
    __builtin_nontemporal_store(valid ? m : 0.0f, out + idx);
}

extern "C" void kernel_launch(void* const* d_in, const int* in_sizes, int n_in,
                              void* d_out, int out_size, void* d_ws, size_t ws_size,
                              hipStream_t stream) {
    const float* feat = (const float*)d_in[0];
    const float* rois = (const float*)d_in[1];
    float* out = (float*)d_out;

    const int R = in_sizes[1] / 5;          // 2048
    const int total = out_size;             // R * 512 * 49
    const int blocks = (total + 255) / 256;
    const size_t need = (size_t)R * 16 * sizeof(int);

    if (d_ws && ws_size >= need) {
        int* bounds = (int*)d_ws;
        roi_setup_kernel<<<(R + 255) / 256, 256, 0, stream>>>(rois, bounds, R);
        roi_pool_kernel<true><<<blocks, 256, 0, stream>>>(feat, rois, bounds, out, total);
    } else {
        roi_pool_kernel<false><<<blocks, 256, 0, stream>>>(feat, rois, (const int*)nullptr, out, total);
    }
}